// Masked_PatchEmbedding_8254927143425
// MI455X (gfx1250) — compile-verified
//
#include <hip/hip_runtime.h>
#include <hip/hip_bf16.h>

typedef __attribute__((ext_vector_type(2))) float v2f;
typedef __attribute__((ext_vector_type(8))) float v8f;

#define TOKN     13824
#define MASKN    10368
#define UNMASKN  3456
#define EMB      768
#define SA_STRIDE 68   // 16x64 A tile padded: bank = (4*row + k) % 64 -> conflict-free

// ---------------------------------------------------------------------------
// Positional-encoding table: penc_ax[pos][c], pos in [0,24), c in [0,256).
// emb[pos, 2j]   = sin(pos * 10000^(-j/128))
// emb[pos, 2j+1] = cos(pos * 10000^(-j/128))
// Full 768-dim enc for token (ph,pw,pd) = concat(tab[ph], tab[pw], tab[pd]).
// ---------------------------------------------------------------------------
__global__ void penc_table_kernel(float* __restrict__ tab) {
    int i = blockIdx.x * blockDim.x + threadIdx.x;
    if (i >= 24 * 256) return;
    int pos = i >> 8, c = i & 255, j = c >> 1;
    float f = __expf(-(float)j * (9.210340371976184f / 128.f)); // ln(10000)/128
    float ang = (float)pos * f;
    tab[i] = (c & 1) ? __cosf(ang) : __sinf(ang);
}

// ---------------------------------------------------------------------------
// Core: 16 output rows x 768 cols per block. 4 waves, each wave: 12 WMMA C
// tiles (192 cols), K=64 via 16 x V_WMMA_F32_16X16X4_F32. Fused bias + posenc
// + LayerNorm + scatter-store.
// ---------------------------------------------------------------------------
template <bool MASKED>
__global__ __launch_bounds__(128)
void mae_embed_kernel(const float* __restrict__ x,        // (4,96,96,96)
                      const float* __restrict__ mtok,     // (10368,64)
                      const float* __restrict__ W,        // (64,768)
                      const float* __restrict__ bias,     // (768)
                      const float* __restrict__ gamma,    // (768)
                      const float* __restrict__ beta,     // (768)
                      const int*   __restrict__ perm,     // (13824)
                      const float* __restrict__ penc,     // (24,256)
                      float* __restrict__ out) {
    __shared__ float sA[16 * SA_STRIDE];
    __shared__ float sPenc[24 * 256];
    __shared__ int   sPos[16 * 3];
    __shared__ float sSum[4][16], sSq[4][16], sMean[16], sRstd[16];

    const int tid = threadIdx.x;
    int batch, r0; // r0: first of this block's 16 rows (mask row or unmask row)
    if (MASKED) { batch = 0;                 r0 = blockIdx.x * 16; }
    else        { batch = blockIdx.x / 216;  r0 = (blockIdx.x % 216) * 16; }

    // --- stage positional-encoding table (24KB) into LDS ---
    for (int i = tid * 4; i < 24 * 256; i += 128 * 4)
        *(float4*)&sPenc[i] = *(const float4*)&penc[i];

    // --- per-row original token index -> (ph, pw, pd) for posenc lookup ---
    if (tid < 16) {
        int idx = MASKED ? perm[r0 + tid] : perm[MASKN + r0 + tid];
        int ph = idx / 576, rem = idx - ph * 576;
        sPos[tid * 3 + 0] = ph;
        sPos[tid * 3 + 1] = rem / 24;
        sPos[tid * 3 + 2] = rem % 24;
    }

    // --- stage A tile (16 rows x 64) into LDS ---
    {
        int trow = tid >> 3;
        int pseg = (tid & 7) * 8;   // patch-dim offset, 8 floats per thread
        if (MASKED) {
            const float* src = mtok + (long)(r0 + trow) * 64 + pseg;
            *(float4*)&sA[trow * SA_STRIDE + pseg]     = *(const float4*)src;
            *(float4*)&sA[trow * SA_STRIDE + pseg + 4] = *(const float4*)(src + 4);
        } else {
            int idx = perm[MASKN + r0 + trow];
            int ph = idx / 576, rem = idx - ph * 576;
            int pw = rem / 24, pd = rem % 24;
            // patch elem p = i*16 + j*4 + k ; x flat = ((b*96+h)*96+w)*96+d
            int i0 = pseg >> 4, j0 = (pseg >> 2) & 3;
            int p1 = pseg + 4;
            int i1 = p1 >> 4,  j1 = (p1 >> 2) & 3;
            long base = (((long)batch * 96 + ph * 4) * 96 + pw * 4) * 96 + pd * 4;
            const float* s0 = x + base + (long)i0 * 9216 + (long)j0 * 96;
            const float* s1 = x + base + (long)i1 * 9216 + (long)j1 * 96;
            *(float4*)&sA[trow * SA_STRIDE + pseg]     = *(const float4*)s0;
            *(float4*)&sA[trow * SA_STRIDE + pseg + 4] = *(const float4*)s1;
        }
    }
    __syncthreads();

    const int wave = tid >> 5, lane = tid & 31;
    const int nl = lane & 15;      // column within 16-wide tile / A row
    const int hg = lane >> 4;      // lane half: selects K pair 0-1 vs 2-3, C rows 0-7 vs 8-15
    const int nb = wave * 192;     // this wave's first column

    v8f acc[12] = {};
    const float* arow = &sA[nl * SA_STRIDE];

    // --- GEMM: 16 K-chunks x 12 N-tiles of V_WMMA_F32_16X16X4_F32 ---
    for (int kc = 0; kc < 16; ++kc) {
        int k0 = kc * 4 + hg * 2;
        v2f a = { arow[k0], arow[k0 + 1] };
        const float* w0 = W + (long)k0 * EMB + nb + nl;
#pragma unroll
        for (int t = 0; t < 12; ++t) {
            v2f b = { w0[t * 16], w0[EMB + t * 16] };
            acc[t] = __builtin_amdgcn_wmma_f32_16x16x4_f32(
                false, a, false, b, (short)0, acc[t], false, false);
        }
    }

    // --- bias + positional encoding, accumulate LayerNorm partials ---
    float s[8] = {}, q[8] = {};
#pragma unroll
    for (int t = 0; t < 12; ++t) {
        int n = nb + t * 16 + nl;
        float bb = bias[n];
        int axis = n >> 8, c = n & 255;
#pragma unroll
        for (int r = 0; r < 8; ++r) {
            int m = r + hg * 8;
            float v = acc[t][r] + bb + sPenc[sPos[m * 3 + axis] * 256 + c];
            acc[t][r] = v;
            s[r] += v;
            q[r] += v * v;
        }
    }
    // butterfly over low 4 lane bits: reduces within each 16-lane group
#pragma unroll
    for (int r = 0; r < 8; ++r) {
#pragma unroll
        for (int off = 1; off < 16; off <<= 1) {
            s[r] += __shfl_xor(s[r], off, 32);
            q[r] += __shfl_xor(q[r], off, 32);
        }
    }
    if (nl == 0) {
#pragma unroll
        for (int r = 0; r < 8; ++r) {
            sSum[wave][hg * 8 + r] = s[r];
            sSq[wave][hg * 8 + r]  = q[r];
        }
    }
    __syncthreads();
    if (tid < 16) {
        float S = 0.f, Q = 0.f;
        for (int w2 = 0; w2 < 4; ++w2) { S += sSum[w2][tid]; Q += sSq[w2][tid]; }
        float mean = S * (1.f / 768.f);
        float var  = Q * (1.f / 768.f) - mean * mean;
        sMean[tid] = mean;
        sRstd[tid] = rsqrtf(var + 0.001f);
    }
    __syncthreads();

    // --- normalize + store ---
#pragma unroll
    for (int t = 0; t < 12; ++t) {
        int n = nb + t * 16 + nl;
        float g = gamma[n], be = beta[n];
#pragma unroll
        for (int r = 0; r < 8; ++r) {
            int m = r + hg * 8;
            float v = (acc[t][r] - sMean[m]) * sRstd[m] * g + be;
            if (MASKED) {
                long tok = (long)(UNMASKN + r0 + m);
#pragma unroll
                for (int b2 = 0; b2 < 4; ++b2)
                    out[((long)b2 * TOKN + tok) * EMB + n] = v;
            } else {
                out[((long)batch * TOKN + r0 + m) * EMB + n] = v;
            }
        }
    }

    // --- mask_idx output (appended after the main tensor, as float) ---
    if (MASKED && tid < 16)
        out[(long)4 * TOKN * EMB + r0 + tid] = (float)perm[r0 + tid];
}

// ---------------------------------------------------------------------------
extern "C" void kernel_launch(void* const* d_in, const int* in_sizes, int n_in,
                              void* d_out, int out_size, void* d_ws, size_t ws_size,
                              hipStream_t stream) {
    const float* x      = (const float*)d_in[0];  // (4,96,96,96,1)
    const float* proj_w = (const float*)d_in[1];  // (64,768)
    const float* proj_b = (const float*)d_in[2];  // (768)
    const float* mtok   = (const float*)d_in[3];  // (1,10368,64)
    const float* gamma  = (const float*)d_in[4];  // (768)
    const float* beta   = (const float*)d_in[5];  // (768)
    const int*   perm   = (const int*)d_in[6];    // (13824)
    float* out  = (float*)d_out;
    float* penc = (float*)d_ws;                   // 24*256 floats = 24KB

    penc_table_kernel<<<24, 256, 0, stream>>>(penc);
    // masked rows: 10368/16 = 648 blocks (broadcast to all 4 batches)
    mae_embed_kernel<true><<<648, 128, 0, stream>>>(
        x, mtok, proj_w, proj_b, gamma, beta, perm, penc, out);
    // unmasked rows: 4 batches * 3456/16 = 864 blocks
    mae_embed_kernel<false><<<864, 128, 0, stream>>>(
        x, mtok, proj_w, proj_b, gamma, beta, perm, penc, out);
}